// PCENTransform_58325655880165
// MI455X (gfx1250) — compile-verified
//
#include <hip/hip_runtime.h>
#include <hip/hip_bf16.h>

typedef __attribute__((ext_vector_type(2))) float v2f;
typedef __attribute__((ext_vector_type(8))) float v8f;

#define EPSC   1e-6f
#define SCOEF  0.025f       // s
#define ACOEF  0.975f       // 1 - s
#define ALPHAC 0.98f
#define DELTAC 2.0f
#define DROOT  1.1892071150027210667f   // 2^0.25

// Problem shape (fixed by the reference)
constexpr int Bn = 64;
constexpr int Tn = 2048;
constexpr int Fn = 128;
constexpr unsigned ROWB = Fn * 4u;     // bytes per (b,t) row

constexpr int NSEG   = 4;            // time segments per sequence
constexpr int SEGLEN = Tn / NSEG;    // 512 steps
constexpr int SEGCH  = SEGLEN / 16;  // 32 output chunks
constexpr int WARMCH = 32;           // 512-step EMA warm-up (a^512 ~ 2.4e-6)

__device__ __forceinline__ float ldf(const char* base, unsigned byteOff) {
    return *(const float*)(base + byteOff);
}

// One wave32 owns (b, 16-channel tile, time segment).
// Chunk of 16 timesteps solved as M = L*X + a_pow (x) carry via 5 chained
// v_wmma_f32_16x16x4_f32, then PCEN epilogue on the D-layout tile.
__global__ __launch_bounds__(256)
void pcen_wmma_kernel(const float* __restrict__ x, float* __restrict__ out) {
    const int lane = threadIdx.x & 31;

    // Wave index is wave-uniform; readfirstlane pins it (and everything
    // derived from it) into SGPRs so tile bases use SADDR addressing.
    const int gw = __builtin_amdgcn_readfirstlane(
        (int)(blockIdx.x * (blockDim.x >> 5) + (threadIdx.x >> 5)));   // 0..2047
    const int seg = gw & (NSEG - 1);
    const int ft  = (gw >> 2) & 7;        // 8 channel tiles of 16
    const int b   = gw >> 5;              // 0..63

    const int half = lane >> 4;           // lane half selects K/M groups
    const int n    = lane & 15;           // row index (A) / col index (B,C,D)

    // a^k table, k = 0..16
    float ap[17];
    ap[0] = 1.0f;
    #pragma unroll
    for (int k = 1; k <= 16; ++k) ap[k] = ap[k - 1] * ACOEF;

    // A-layout (16x4 f32): lane m=n holds K = 4*s + 2*half + v in VGPR v.
    // L[m][K] = s * a^(m-K) for m >= K else 0.
    v2f Ls[4];
    #pragma unroll
    for (int s = 0; s < 4; ++s) {
        #pragma unroll
        for (int v = 0; v < 2; ++v) {
            const int K = 4 * s + 2 * half + v;
            float val = (n >= K) ? (SCOEF * ap[n - K]) : 0.0f;
            if (v == 0) Ls[s].x = val; else Ls[s].y = val;
        }
    }
    // Carry matrix column: A_c[m][0] = a^(m+1); other K columns zero.
    v2f Ac;
    Ac.x = (half == 0) ? ap[n + 1] : 0.0f;   // K=0 lives in lanes 0-15, VGPR0
    Ac.y = 0.0f;

    // Uniform (SGPR) tile bases; per-lane state is a 32-bit BYTE offset so
    // compile-time row displacements fold into the 24-bit instruction offset.
    const char* __restrict__ xb8 = (const char*)(x   + (size_t)b * Tn * Fn + ft * 16);
    char* __restrict__       ob8 = (char*)      (out + (size_t)b * Tn * Fn + ft * 16);

    const int t_start = seg * SEGLEN;
    const int warm    = (seg == 0) ? 0 : WARMCH;
    const int tfirst  = t_start - warm * 16;

    // B-layout loads: rows (4s + v) relative; half contributes +2 rows.
    unsigned offB = (unsigned)(tfirst + 2 * half) * ROWB + (unsigned)n * 4u;
    // D-layout loads/stores: rows v relative; half contributes +8 rows.
    unsigned offD = (unsigned)(t_start + 8 * half) * ROWB + (unsigned)n * 4u;
    // Per-lane prefetch row, one chunk ahead.
    unsigned offP = (unsigned)(tfirst + 16 + lane) * ROWB + (unsigned)n * 4u;

    // Seed carry: since a = 1-s, c = x[tfirst] makes m[tfirst] = x[tfirst].
    float c = ldf(xb8, (unsigned)tfirst * ROWB + (unsigned)n * 4u);

    for (int ch = -warm; ch < SEGCH; ++ch) {
        const int tc = t_start + ch * 16;

        // Speculative prefetch of next chunk rows (global_prefetch_b8).
        if (tc + 16 + lane < Tn) {
            __builtin_prefetch(xb8 + offP, 0, 1);
        }
        offP += 16u * ROWB;

        v8f acc = {};

        // Carry term: rank-1 update  a_pow (16x1) * c (1x16)  as one WMMA.
        {
            v2f Bc;
            Bc.x = (half == 0) ? c : 0.0f;   // B row K=0: lanes 0-15, VGPR0
            Bc.y = 0.0f;
            acc = __builtin_amdgcn_wmma_f32_16x16x4_f32(
                false, Ac, false, Bc, (short)0, acc, false, false);
        }

        // D += L[:,4s:4s+4] * X[4s:4s+4, :]  for the 4 K-slices.
        #pragma unroll
        for (int s = 0; s < 4; ++s) {
            v2f Bx;  // B-layout 4x16: lane n holds rows 2*half+v of the slice
            Bx.x = ldf(xb8, offB + (unsigned)(4 * s + 0) * ROWB);
            Bx.y = ldf(xb8, offB + (unsigned)(4 * s + 1) * ROWB);
            acc = __builtin_amdgcn_wmma_f32_16x16x4_f32(
                false, Ls[s], false, Bx, (short)0, acc, false, false);
        }
        offB += 16u * ROWB;

        // New carry = M[15][n]  (D VGPR7, lanes 16-31).
        c = __shfl(acc[7], 16 + n, 32);

        if (ch >= 0) {
            // PCEN epilogue on the D-layout tile: row = tc + v + 8*half.
            #pragma unroll
            for (int v = 0; v < 8; ++v) {
                const unsigned o = offD + (unsigned)v * ROWB;
                const float xv = ldf(xb8, o);
                const float mm = acc[v] + EPSC;
                // x / (M+eps)^alpha  =  x * exp2(-alpha * log2(M+eps))
                const float ratio =
                    xv * __builtin_amdgcn_exp2f(-ALPHAC * __builtin_amdgcn_logf(mm));
                // (ratio + delta)^0.25 = sqrt(sqrt(.))
                const float val =
                    __builtin_amdgcn_sqrtf(__builtin_amdgcn_sqrtf(ratio + DELTAC)) - DROOT;
                // Streamed write-once output: non-temporal store.
                __builtin_nontemporal_store(val, (float*)(ob8 + o));
            }
            offD += 16u * ROWB;
        }
    }
}

extern "C" void kernel_launch(void* const* d_in, const int* in_sizes, int n_in,
                              void* d_out, int out_size, void* d_ws, size_t ws_size,
                              hipStream_t stream) {
    (void)in_sizes; (void)n_in; (void)d_ws; (void)ws_size; (void)out_size;
    const float* x = (const float*)d_in[0];
    float* out = (float*)d_out;

    // 2048 waves total: 64 b * 8 channel-tiles * 4 time segments.
    // 256 threads (8 waves) per block -> 256 blocks.
    const int totalWaves = Bn * (Fn / 16) * NSEG;
    const int wavesPerBlock = 8;
    dim3 grid(totalWaves / wavesPerBlock);
    dim3 block(wavesPerBlock * 32);
    pcen_wmma_kernel<<<grid, block, 0, stream>>>(x, out);
}